// SC2MambaBridge_57045755625695
// MI455X (gfx1250) — compile-verified
//
#include <hip/hip_runtime.h>
#include <hip/hip_bf16.h>
#include <math.h>

// ---------------------------------------------------------------------------
// SC2 Mamba bridge forward for gfx1250 (MI455X), wave32, WMMA fp32 path.
//   N_GENES=2000, D_MODEL=256, N_LAYERS=4, D_STATE=16, D_CONV=4,
//   D_INNER=512, DT_RANK=16, B=2, M = B*L = 4000 rows.
// ---------------------------------------------------------------------------

typedef __attribute__((ext_vector_type(2))) float v2f;
typedef __attribute__((ext_vector_type(8))) float v8f;

#define BATCH   2
#define SEQLEN  2000
#define NROWS   (BATCH * SEQLEN)   // 4000
#define DMODEL  256
#define DINNER  512
#define DSTATE  16
#define DTRANK  16
#define XDBC_W  (DTRANK + 2 * DSTATE)  // 48

static __device__ __forceinline__ float softplusf(float x) {
    return (x > 20.0f) ? x : log1pf(expf(x));
}
static __device__ __forceinline__ float siluf(float x) {
    return x / (1.0f + expf(-x));
}

// ---------------------------------------------------------------------------
// Generic WMMA GEMM:  C[M x N] = act( A[M x K] * W[N x K]^T + bias ) (+= C)
// One 16x16 output tile per wave (wave32). K must be a multiple of 4,
// N a multiple of 16. Tiles beyond M are rejected per whole wave (threadIdx.y
// uniform), so EXEC is all-ones whenever v_wmma executes.
//
// Fragment layouts (CDNA5 ISA 7.12.2, 32-bit):
//   A 16x4 :  lane (g=lane>>4, r=lane&15) vgpr j  ->  A[r][2g + j]
//   B 4x16 :  lane (g, r)                vgpr j  ->  B[2g + j][r] = W[r][2g+j]
//   D 16x16:  lane (g, r)                vgpr v  ->  D[v + 8g][r]
// ---------------------------------------------------------------------------
__global__ void gemm_wmma_nt(const float* __restrict__ A, int lda,
                             const float* __restrict__ W, int ldw,
                             float* __restrict__ C, int ldc,
                             int M, int N, int K,
                             const float* __restrict__ bias,
                             int act,     // 0 = none, 1 = softplus
                             int addC)    // 1 = C += result (residual)
{
    const int lane = threadIdx.x;                       // 0..31
    const int tn   = blockIdx.x;                        // tile col
    const int tm   = blockIdx.y * blockDim.y + threadIdx.y; // tile row
    if (tm * 16 >= M) return;                           // uniform per wave

    const int r = lane & 15;
    const int g = lane >> 4;

    const float* __restrict__ Arow = A + (size_t)(tm * 16 + r) * lda + 2 * g;
    const float* __restrict__ Wrow = W + (size_t)(tn * 16 + r) * ldw + 2 * g;

    v8f acc = {};
    for (int k = 0; k < K; k += 4) {
        v2f a, b;
        a[0] = Arow[k];  a[1] = Arow[k + 1];
        b[0] = Wrow[k];  b[1] = Wrow[k + 1];
        acc = __builtin_amdgcn_wmma_f32_16x16x4_f32(
                  false, a, false, b, (short)0, acc, false, false);
    }

    const int ncol = tn * 16 + r;
    const float bv = bias ? bias[ncol] : 0.0f;
#pragma unroll
    for (int v = 0; v < 8; ++v) {
        const int row = tm * 16 + v + 8 * g;
        float val = acc[v] + bv;
        if (act == 1) val = softplusf(val);
        float* cp = &C[(size_t)row * ldc + ncol];
        if (addC) val += *cp;
        *cp = val;
    }
}

// ---------------------------------------------------------------------------
// h[b,l,d] = x[b,l]*bulk_in_w[d] + bulk_in_b[d] + gene_emb[l,d] + mod_emb[d]
// ---------------------------------------------------------------------------
__global__ void embed_kernel(const float* __restrict__ x,
                             const float* __restrict__ bw,
                             const float* __restrict__ bb,
                             const float* __restrict__ ge,
                             const float* __restrict__ me,
                             float* __restrict__ h, int total)
{
    int i = blockIdx.x * blockDim.x + threadIdx.x;
    if (i >= total) return;
    int d   = i & (DMODEL - 1);
    int row = i >> 8;            // b*L + l
    int l   = row % SEQLEN;
    h[i] = x[row] * bw[d] + bb[d] + ge[l * DMODEL + d] + me[d];
}

// ---------------------------------------------------------------------------
// Wave-per-row LayerNorm over DMODEL=256 (8 elements / lane, shfl reduce).
// ---------------------------------------------------------------------------
__global__ void ln_kernel(const float* __restrict__ X,
                          const float* __restrict__ w,
                          const float* __restrict__ b,
                          float* __restrict__ Y, int M)
{
    const int lane = threadIdx.x & 31;
    const int wid  = threadIdx.x >> 5;
    const int row  = blockIdx.x * 8 + wid;
    if (row >= M) return;
    const float* xr = X + (size_t)row * DMODEL;

    float v[8];
    float s = 0.0f;
#pragma unroll
    for (int i = 0; i < 8; ++i) { v[i] = xr[lane + 32 * i]; s += v[i]; }
#pragma unroll
    for (int off = 16; off; off >>= 1) s += __shfl_xor(s, off, 32);
    const float mu = s * (1.0f / DMODEL);

    float vs = 0.0f;
#pragma unroll
    for (int i = 0; i < 8; ++i) { float d = v[i] - mu; vs += d * d; }
#pragma unroll
    for (int off = 16; off; off >>= 1) vs += __shfl_xor(vs, off, 32);
    const float inv = rsqrtf(vs * (1.0f / DMODEL) + 1e-5f);

    float* yr = Y + (size_t)row * DMODEL;
#pragma unroll
    for (int i = 0; i < 8; ++i) {
        int d = lane + 32 * i;
        yr[d] = (v[i] - mu) * inv * w[d] + b[d];
    }
}

// ---------------------------------------------------------------------------
// Causal depthwise conv (D_CONV=4) over the u-half of xz, + bias, + SiLU.
// ---------------------------------------------------------------------------
__global__ void conv_silu_kernel(const float* __restrict__ xz,
                                 const float* __restrict__ cw,
                                 const float* __restrict__ cb,
                                 float* __restrict__ u, int total)
{
    int i = blockIdx.x * blockDim.x + threadIdx.x;
    if (i >= total) return;
    int e   = i & (DINNER - 1);
    int row = i >> 9;            // b*L + l
    int l   = row % SEQLEN;
    float acc = cb[e];
#pragma unroll
    for (int k = 0; k < 4; ++k) {
        int ll = l + k - 3;
        if (ll >= 0)
            acc += xz[(size_t)(row + k - 3) * (2 * DINNER) + e] * cw[e * 4 + k];
    }
    u[i] = siluf(acc);
}

// ---------------------------------------------------------------------------
// Selective scan: one thread per (b, d) channel, 16-state in registers.
// B/C (16 floats each per timestep) staged in LDS per block.
// Fused: y = (sum_n h_n * C_n + u*D) * silu(z).
// ---------------------------------------------------------------------------
__global__ void scan_kernel(const float* __restrict__ dt,
                            const float* __restrict__ u,
                            const float* __restrict__ xdbc,
                            const float* __restrict__ xz,
                            const float* __restrict__ A_log,
                            const float* __restrict__ Dv,
                            float* __restrict__ ys)
{
    __shared__ float sBC[2 * DSTATE];
    const int d = blockIdx.x * blockDim.x + threadIdx.x;  // 0..511
    const int b = blockIdx.y;

    float An[DSTATE], st[DSTATE];
#pragma unroll
    for (int n = 0; n < DSTATE; ++n) {
        An[n] = -expf(A_log[d * DSTATE + n]);
        st[n] = 0.0f;
    }
    const float Dd = Dv[d];

    for (int l = 0; l < SEQLEN; ++l) {
        const size_t row = (size_t)b * SEQLEN + l;
        __syncthreads();
        if (threadIdx.x < 2 * DSTATE)
            sBC[threadIdx.x] = xdbc[row * XDBC_W + DTRANK + threadIdx.x];
        __syncthreads();

        const float dtv = dt[row * DINNER + d];
        const float uv  = u[row * DINNER + d];
        const float dbu = dtv * uv;
        float acc = 0.0f;
#pragma unroll
        for (int n = 0; n < DSTATE; ++n) {
            st[n] = expf(dtv * An[n]) * st[n] + dbu * sBC[n];
            acc += st[n] * sBC[DSTATE + n];
        }
        const float zv = xz[row * (2 * DINNER) + DINNER + d];
        ys[row * DINNER + d] = (acc + uv * Dd) * siluf(zv);
    }
}

// ---------------------------------------------------------------------------
// Head: y[row] = dot(hn[row,:], head_w) + head_b   (wave per row)
// ---------------------------------------------------------------------------
__global__ void head_kernel(const float* __restrict__ X,
                            const float* __restrict__ hw,
                            const float* __restrict__ hb,
                            float* __restrict__ out, int M)
{
    const int lane = threadIdx.x & 31;
    const int wid  = threadIdx.x >> 5;
    const int row  = blockIdx.x * 8 + wid;
    if (row >= M) return;
    float s = 0.0f;
#pragma unroll
    for (int i = 0; i < 8; ++i) {
        int d = lane + 32 * i;
        s += X[(size_t)row * DMODEL + d] * hw[d];
    }
#pragma unroll
    for (int off = 16; off; off >>= 1) s += __shfl_xor(s, off, 32);
    if (lane == 0) out[row] = s + hb[0];
}

// ---------------------------------------------------------------------------
// Host-side orchestration
// ---------------------------------------------------------------------------
extern "C" void kernel_launch(void* const* d_in, const int* in_sizes, int n_in,
                              void* d_out, int out_size, void* d_ws, size_t ws_size,
                              hipStream_t stream)
{
    const float* x        = (const float*)d_in[0];   // (2,2000)
    const float* bulk_w   = (const float*)d_in[1];   // (256)
    const float* bulk_b   = (const float*)d_in[2];   // (256)
    const float* gene_emb = (const float*)d_in[3];   // (1,2000,256)
    const float* mod_emb  = (const float*)d_in[4];   // (1,1,256)
    const float* ln_w     = (const float*)d_in[5];   // (4,256)
    const float* ln_b     = (const float*)d_in[6];   // (4,256)
    const float* in_w     = (const float*)d_in[7];   // (4,1024,256)
    const float* conv_w   = (const float*)d_in[8];   // (4,512,4)
    const float* conv_b   = (const float*)d_in[9];   // (4,512)
    const float* xp_w     = (const float*)d_in[10];  // (4,48,512)
    const float* dtp_w    = (const float*)d_in[11];  // (4,512,16)
    const float* dtp_b    = (const float*)d_in[12];  // (4,512)
    const float* A_log    = (const float*)d_in[13];  // (4,512,16)
    const float* Dvec     = (const float*)d_in[14];  // (4,512)
    const float* out_w    = (const float*)d_in[15];  // (4,256,512)
    const float* fin_w    = (const float*)d_in[16];  // (256)
    const float* fin_b    = (const float*)d_in[17];  // (256)
    const float* head_w   = (const float*)d_in[18];  // (256)
    const float* head_b   = (const float*)d_in[19];  // (1)
    float* out = (float*)d_out;                      // (2,2000) fp32

    // Workspace carve-out (~50 MB of fp32)
    float* ws   = (float*)d_ws;
    float* h    = ws;  ws += (size_t)NROWS * DMODEL;       // residual stream
    float* hn   = ws;  ws += (size_t)NROWS * DMODEL;       // layernorm out
    float* xz   = ws;  ws += (size_t)NROWS * 2 * DINNER;   // in_proj out
    float* u    = ws;  ws += (size_t)NROWS * DINNER;       // conv+silu out
    float* xdbc = ws;  ws += (size_t)NROWS * XDBC_W;       // x_proj out
    float* dt   = ws;  ws += (size_t)NROWS * DINNER;       // softplus(dt_proj)
    float* ys   = ws;  ws += (size_t)NROWS * DINNER;       // scan out (gated)

    const int M = NROWS;                       // 4000 rows, 250 row tiles
    const dim3 gblk(32, 4);                    // 4 waves / block
    const int  gy = (M / 16 + 3) / 4;          // 63 (tile guard in kernel)

    // Embedding
    embed_kernel<<<(M * DMODEL + 255) / 256, 256, 0, stream>>>(
        x, bulk_w, bulk_b, gene_emb, mod_emb, h, M * DMODEL);

    for (int i = 0; i < 4; ++i) {
        // LayerNorm
        ln_kernel<<<(M + 7) / 8, 256, 0, stream>>>(
            h, ln_w + i * DMODEL, ln_b + i * DMODEL, hn, M);

        // in_proj: xz = hn @ in_w^T   (N=1024, K=256)
        gemm_wmma_nt<<<dim3(2 * DINNER / 16, gy), gblk, 0, stream>>>(
            hn, DMODEL, in_w + (size_t)i * 2 * DINNER * DMODEL, DMODEL,
            xz, 2 * DINNER, M, 2 * DINNER, DMODEL, nullptr, 0, 0);

        // depthwise conv + bias + SiLU on u-half
        conv_silu_kernel<<<(M * DINNER + 255) / 256, 256, 0, stream>>>(
            xz, conv_w + (size_t)i * DINNER * 4, conv_b + (size_t)i * DINNER,
            u, M * DINNER);

        // x_proj: xdbc = u @ xp_w^T   (N=48, K=512)
        gemm_wmma_nt<<<dim3(XDBC_W / 16, gy), gblk, 0, stream>>>(
            u, DINNER, xp_w + (size_t)i * XDBC_W * DINNER, DINNER,
            xdbc, XDBC_W, M, XDBC_W, DINNER, nullptr, 0, 0);

        // dt_proj + bias + softplus: dt = softplus(dt_r @ dtp_w^T + dtp_b)
        // A is the first DTRANK columns of xdbc (lda = 48).
        gemm_wmma_nt<<<dim3(DINNER / 16, gy), gblk, 0, stream>>>(
            xdbc, XDBC_W, dtp_w + (size_t)i * DINNER * DTRANK, DTRANK,
            dt, DINNER, M, DINNER, DTRANK, dtp_b + (size_t)i * DINNER, 1, 0);

        // selective scan + skip + SiLU(z) gate
        scan_kernel<<<dim3(DINNER / 256, BATCH), 256, 0, stream>>>(
            dt, u, xdbc, xz,
            A_log + (size_t)i * DINNER * DSTATE, Dvec + (size_t)i * DINNER, ys);

        // out_proj with residual accumulate: h += ys @ out_w^T (N=256, K=512)
        gemm_wmma_nt<<<dim3(DMODEL / 16, gy), gblk, 0, stream>>>(
            ys, DINNER, out_w + (size_t)i * DMODEL * DINNER, DINNER,
            h, DMODEL, M, DMODEL, DINNER, nullptr, 0, 1);
    }

    // Final LayerNorm + head
    ln_kernel<<<(M + 7) / 8, 256, 0, stream>>>(h, fin_w, fin_b, hn, M);
    head_kernel<<<(M + 7) / 8, 256, 0, stream>>>(hn, head_w, head_b, out, M);
}